// MambaBlock_3238405341854
// MI455X (gfx1250) — compile-verified
//
#include <hip/hip_runtime.h>
#include <hip/hip_bf16.h>

// ---------------- problem constants (match reference) ----------------
#define BB        2
#define LL        1024
#define DMODEL    1024
#define DINNER    2048          // EXPAND * DMODEL
#define DSTATE    16
#define DCONV     4
#define DTRANK    64
#define XDBL      (DTRANK + 2 * DSTATE)   // 96
#define LN_EPS    1e-5f

#define NSPLIT    4                       // lanes per (b,d) in the scan
#define SPERL     (DSTATE / NSPLIT)       // states per lane = 4
#define KC        64                      // K-chunk staged to LDS per TDM op

typedef float v2f __attribute__((ext_vector_type(2)));
typedef float v4f __attribute__((ext_vector_type(4)));
typedef float v8f __attribute__((ext_vector_type(8)));
typedef unsigned int u32x4 __attribute__((ext_vector_type(4)));
typedef int i32x4 __attribute__((ext_vector_type(4)));
typedef int i32x8 __attribute__((ext_vector_type(8)));

// =====================================================================
// TDM: DMA one B panel tile (KC rows x W cols, fp32, row stride ldb) from
// global to LDS, padding +4 dwords per row (bank-conflict-free stride W+4).
// D# per CDNA5 ISA 8.3/8.4.  W must be 32 or 64 (pad_interval = W dwords).
// =====================================================================
template <int W>
__device__ __forceinline__ void tdm_load_b_tile(const float* gptr, int ldb,
                                                unsigned lds_addr)
{
    const unsigned long long ga = (unsigned long long)(uintptr_t)gptr;
    // group0: count=1 | lds_addr | global_addr[56:0] | type=2
    u32x4 g0 = { 1u, lds_addr, (unsigned)ga,
                 (unsigned)(ga >> 32) | 0x80000000u };
    // group1
    const unsigned pad_interval = (W == 64) ? 5u : 4u;   // 64 / 32 dwords
    const int d0 = (int)((2u << 16)            // data_size = 4 bytes
                       | (1u << 20)            // pad_enable
                       | (pad_interval << 22)  // pad every row
                       | (3u << 25));          // pad_amount = 4 dwords
    const int d1 = (int)((unsigned)W << 16);            // tensor_dim0 = W
    const int d2 = (int)((unsigned)KC << 16);           // tensor_dim1 = KC
    const int d3 = (int)((unsigned)W << 16);            // tile_dim0 = W
    const int d4 = (int)KC;                             // tile_dim1 = KC
    const int d5 = (int)ldb;                            // dim0 stride lo
    i32x8 g1 = { d0, d1, d2, d3, d4, d5, 0, 0 };
    i32x4 gz = { 0, 0, 0, 0 };
#if __clang_major__ >= 23
    i32x8 gz8 = { 0, 0, 0, 0, 0, 0, 0, 0 };
    __builtin_amdgcn_tensor_load_to_lds(g0, g1, gz, gz, gz8, 0);
#else
    __builtin_amdgcn_tensor_load_to_lds(g0, g1, gz, gz, 0);
#endif
}

// =====================================================================
// fp32 GEMM: C[M,N] = A[M,K] * B[K,N] via V_WMMA_F32_16X16X4_F32.
// Wave32; 8 waves/WG stacked in M, each wave owns a 32 x (NT*16) tile
// (MT=2 row-tiles).  The shared B panel is staged to LDS by the Tensor
// Data Mover, double-buffered (wave0 issues DMA, s_wait_tensorcnt +
// barrier publishes it).  A fragments stream from global (L2-resident).
// Requires M % 256 == 0, K % KC == 0, n_off + gridDim.x*NT*16 <= N.
// =====================================================================
template <int NT>
__global__ __launch_bounds__(256) void gemm_f32_wmma_kernel(
    const float* __restrict__ A, int lda,
    const float* __restrict__ Bm, int ldb,
    float* __restrict__ C, int ldc,
    int K, int n_off)
{
    constexpr int W  = NT * 16;       // panel width
    constexpr int WS = W + 4;         // padded LDS row stride (floats)
    __shared__ float bsm[2][WS * KC];

    const int wave = threadIdx.x >> 5;     // 0..7
    const int lane = threadIdx.x & 31;
    const int half = lane >> 4;            // 0 or 1
    const int r    = lane & 15;            // 0..15

    const int m0     = (blockIdx.y * 8 + wave) * 32;    // 2 row-tiles
    const int n_base = n_off + blockIdx.x * W;

    const v8f vzero = {0.f, 0.f, 0.f, 0.f, 0.f, 0.f, 0.f, 0.f};
    v8f acc0[NT], acc1[NT];
#pragma unroll
    for (int t = 0; t < NT; ++t) { acc0[t] = vzero; acc1[t] = vzero; }

    const size_t arow0 = (size_t)(m0 + r)      * (size_t)lda;
    const size_t arow1 = (size_t)(m0 + 16 + r) * (size_t)lda;

    const int NCHUNK = K / KC;

    // prime the pipeline: chunk 0 -> buffer 0
    if (wave == 0)
        tdm_load_b_tile<W>(Bm + (size_t)0 * ldb + n_base, ldb,
                           (unsigned)(uintptr_t)&bsm[0][0]);

    for (int c = 0; c < NCHUNK; ++c) {
        if (wave == 0) {
            if (c + 1 < NCHUNK) {
                tdm_load_b_tile<W>(
                    Bm + (size_t)(c + 1) * KC * ldb + n_base, ldb,
                    (unsigned)(uintptr_t)&bsm[(c + 1) & 1][0]);
                __builtin_amdgcn_s_wait_tensorcnt(1);   // chunk c landed
            } else {
                __builtin_amdgcn_s_wait_tensorcnt(0);
            }
        }
        __syncthreads();                                // publish chunk c

        if (c + 1 < NCHUNK) {                           // warm L2/L1 for next A
            __builtin_prefetch(A + arow0 + (size_t)(c + 1) * KC, 0, 3);
            __builtin_prefetch(A + arow1 + (size_t)(c + 1) * KC, 0, 3);
        }

        const float* bs = &bsm[c & 1][0];
        const size_t kbase = (size_t)c * KC;

#pragma unroll 4
        for (int kk = 0; kk < KC; kk += 4) {
            // A fragments (16x4): v0 lanes0-15 K=k, lanes16-31 K=k+2; v1 +1
            const size_t ak = kbase + (size_t)(kk + 2 * half);
            const v2f a0 = *reinterpret_cast<const v2f*>(A + arow0 + ak);
            const v2f a1 = *reinterpret_cast<const v2f*>(A + arow1 + ak);

            // B fragments from LDS (padded stride kills bank conflicts)
            const int brow0 = (kk + 2 * half) * WS;
            v2f b[NT];
#pragma unroll
            for (int t = 0; t < NT; ++t) {
                b[t].x = bs[brow0 +      t * 16 + r];
                b[t].y = bs[brow0 + WS + t * 16 + r];
            }
#pragma unroll
            for (int t = 0; t < NT; ++t) {
                acc0[t] = __builtin_amdgcn_wmma_f32_16x16x4_f32(
                              false, a0, false, b[t], (short)0, acc0[t],
                              false, false);
                acc1[t] = __builtin_amdgcn_wmma_f32_16x16x4_f32(
                              false, a1, false, b[t], (short)0, acc1[t],
                              false, false);
            }
        }
        __syncthreads();    // all waves done with buffer (c&1) before reuse
    }

    // D layout: VGPR v, lanes0-15 -> M=base+v, lanes16-31 -> M=base+v+8.
#pragma unroll
    for (int t = 0; t < NT; ++t) {
        const int n0 = n_base + t * 16;
#pragma unroll
        for (int v = 0; v < 8; ++v) {
            C[(size_t)(m0 + v + 8 * half)      * (size_t)ldc + n0 + r] = acc0[t][v];
            C[(size_t)(m0 + 16 + v + 8 * half) * (size_t)ldc + n0 + r] = acc1[t][v];
        }
    }
}

// =====================================================================
// Depthwise causal conv1d (K=4, left pad 3) + bias + SiLU.
// =====================================================================
__global__ __launch_bounds__(256) void conv_silu_kernel(
    const float* __restrict__ xz, const float* __restrict__ conv_w,
    const float* __restrict__ conv_b, float* __restrict__ xc)
{
    const int idx = blockIdx.x * blockDim.x + threadIdx.x;   // over B*L*DINNER
    if (idx >= BB * LL * DINNER) return;
    const int d = idx % DINNER;
    const int l = (idx / DINNER) % LL;
    const int b = idx / (DINNER * LL);

    float s = conv_b[d];
#pragma unroll
    for (int j = 0; j < DCONV; ++j) {
        const int t = l - (DCONV - 1) + j;
        if (t >= 0)
            s += xz[((size_t)(b * LL + t)) * (2 * DINNER) + d] * conv_w[d * DCONV + j];
    }
    xc[idx] = s / (1.0f + __expf(-s));    // SiLU
}

// =====================================================================
// delta = softplus(dtproj + b_dt)   (in place)
// =====================================================================
__global__ __launch_bounds__(256) void softplus_bias_kernel(
    float* __restrict__ delta, const float* __restrict__ b_dt)
{
    const int idx = blockIdx.x * blockDim.x + threadIdx.x;   // over B*L*DINNER
    if (idx >= BB * LL * DINNER) return;
    const float v = delta[idx] + b_dt[idx % DINNER];
    delta[idx] = (v > 20.0f) ? v : log1pf(__expf(v));
}

// =====================================================================
// Selective scan over L. 4 lanes per (b,d) channel, 4 states each;
// per-step y combined via shfl_xor.  Fuses +xc*D and *silu(z).
// =====================================================================
__global__ __launch_bounds__(256) void scan_kernel(
    const float* __restrict__ delta, const float* __restrict__ xc,
    const float* __restrict__ x_dbl, const float* __restrict__ xz,
    const float* __restrict__ A_log, const float* __restrict__ D_param,
    float* __restrict__ ygated)
{
    const int idx = blockIdx.x * blockDim.x + threadIdx.x;   // B*DINNER*NSPLIT
    if (idx >= BB * DINNER * NSPLIT) return;
    const int sg = idx & (NSPLIT - 1);        // state group 0..3
    const int d  = (idx / NSPLIT) % DINNER;
    const int b  = idx / (NSPLIT * DINNER);

    float Arow[SPERL], h[SPERL];
#pragma unroll
    for (int j = 0; j < SPERL; ++j) {
        Arow[j] = -__expf(A_log[d * DSTATE + sg * SPERL + j]);
        h[j] = 0.0f;
    }
    const float Dp = D_param[d];

    for (int l = 0; l < LL; ++l) {
        const size_t row = (size_t)(b * LL + l);
        const float dlt = delta[row * DINNER + d];
        const float u   = xc[row * DINNER + d];
        const float du  = dlt * u;

        const v4f Bv = *reinterpret_cast<const v4f*>(
            x_dbl + row * XDBL + DTRANK + sg * SPERL);
        const v4f Cv = *reinterpret_cast<const v4f*>(
            x_dbl + row * XDBL + DTRANK + DSTATE + sg * SPERL);

        float y = 0.0f;
#pragma unroll
        for (int j = 0; j < SPERL; ++j) {
            const float dA = __expf(dlt * Arow[j]);
            h[j] = dA * h[j] + du * Bv[j];
            y += h[j] * Cv[j];
        }
        y += __shfl_xor(y, 1, 32);
        y += __shfl_xor(y, 2, 32);

        if (sg == 0) {
            y += u * Dp;
            const float z = xz[row * (2 * DINNER) + DINNER + d];
            y *= z / (1.0f + __expf(-z));     // * silu(z)
            ygated[row * DINNER + d] = y;
        }
    }
}

// =====================================================================
// LayerNorm over last dim (DMODEL), one workgroup (8 waves) per row.
// =====================================================================
__global__ __launch_bounds__(256) void layernorm_kernel(
    const float* __restrict__ in, const float* __restrict__ g,
    const float* __restrict__ beta, float* __restrict__ out)
{
    __shared__ float red_s[8];
    __shared__ float red_ss[8];
    __shared__ float s_mu, s_rstd;

    const int row = blockIdx.x;                 // 0 .. B*L-1
    const float* p = in + (size_t)row * DMODEL;

    float s = 0.0f, ss = 0.0f;
    for (int i = threadIdx.x; i < DMODEL; i += 256) {
        const float v = p[i];
        s += v; ss += v * v;
    }
#pragma unroll
    for (int off = 16; off >= 1; off >>= 1) {
        s  += __shfl_xor(s,  off, 32);
        ss += __shfl_xor(ss, off, 32);
    }
    const int wave = threadIdx.x >> 5, lane = threadIdx.x & 31;
    if (lane == 0) { red_s[wave] = s; red_ss[wave] = ss; }
    __syncthreads();
    if (threadIdx.x == 0) {
        float ts = 0.0f, tss = 0.0f;
#pragma unroll
        for (int w = 0; w < 8; ++w) { ts += red_s[w]; tss += red_ss[w]; }
        const float m = ts / (float)DMODEL;
        s_mu = m;
        s_rstd = rsqrtf(tss / (float)DMODEL - m * m + LN_EPS);
    }
    __syncthreads();
    const float m = s_mu, r = s_rstd;
    for (int i = threadIdx.x; i < DMODEL; i += 256)
        out[(size_t)row * DMODEL + i] = (p[i] - m) * r * g[i] + beta[i];
}

// =====================================================================
// Host-side orchestration
// =====================================================================
extern "C" void kernel_launch(void* const* d_in, const int* in_sizes, int n_in,
                              void* d_out, int out_size, void* d_ws, size_t ws_size,
                              hipStream_t stream) {
    (void)in_sizes; (void)n_in; (void)out_size; (void)ws_size;

    const float* x       = (const float*)d_in[0];
    const float* W_in    = (const float*)d_in[1];
    const float* conv_w  = (const float*)d_in[2];
    const float* conv_b  = (const float*)d_in[3];
    const float* W_x     = (const float*)d_in[4];
    const float* W_dt    = (const float*)d_in[5];
    const float* b_dt    = (const float*)d_in[6];
    const float* A_log   = (const float*)d_in[7];
    const float* D_param = (const float*)d_in[8];
    const float* W_out   = (const float*)d_in[9];
    const float* ln_g    = (const float*)d_in[10];
    const float* ln_b    = (const float*)d_in[11];
    float* out = (float*)d_out;

    // workspace layout (floats)
    float* ws     = (float*)d_ws;
    const size_t M = (size_t)BB * LL;            // 2048 rows
    float* xz     = ws;                          // [M, 2*DINNER]
    float* xc     = xz    + M * (2 * DINNER);    // [M, DINNER]
    float* x_dbl  = xc    + M * DINNER;          // [M, 96]
    float* deltab = x_dbl + M * XDBL;            // [M, DINNER]
    float* ygated = deltab + M * DINNER;         // [M, DINNER]
    float* preln  = ygated + M * DINNER;         // [M, DMODEL]

    const int MGRID = (int)M / 256;              // 8 workgroup rows
    dim3 blk(256);

    // 1) xz = x @ W_in            (2048 x 1024 x 4096)
    gemm_f32_wmma_kernel<4><<<dim3((2 * DINNER) / 64, MGRID), blk, 0, stream>>>(
        x, DMODEL, W_in, 2 * DINNER, xz, 2 * DINNER, DMODEL, 0);

    // 2) xc = silu(causal_dwconv(xp) + conv_b)
    conv_silu_kernel<<<(BB * LL * DINNER + 255) / 256, blk, 0, stream>>>(
        xz, conv_w, conv_b, xc);

    // 3) x_dbl = xc @ W_x         (2048 x 2048 x 96) -> [dt | B | C]
    gemm_f32_wmma_kernel<4><<<dim3(1, MGRID), blk, 0, stream>>>(
        xc, DINNER, W_x, XDBL, x_dbl, XDBL, DINNER, 0);
    gemm_f32_wmma_kernel<2><<<dim3(1, MGRID), blk, 0, stream>>>(
        xc, DINNER, W_x, XDBL, x_dbl, XDBL, DINNER, 64);

    // 4) deltab = dt @ W_dt       (2048 x 64 x 2048), dt = x_dbl[:, :64]
    gemm_f32_wmma_kernel<4><<<dim3(DINNER / 64, MGRID), blk, 0, stream>>>(
        x_dbl, XDBL, W_dt, DINNER, deltab, DINNER, DTRANK, 0);

    //    delta = softplus(deltab + b_dt)
    softplus_bias_kernel<<<(BB * LL * DINNER + 255) / 256, blk, 0, stream>>>(
        deltab, b_dt);

    // 5) selective scan + skip + gate (4 lanes per channel)
    scan_kernel<<<(BB * DINNER * NSPLIT + 255) / 256, blk, 0, stream>>>(
        deltab, xc, x_dbl, xz, A_log, D_param, ygated);

    // 6) preln = ygated @ W_out   (2048 x 2048 x 1024)
    gemm_f32_wmma_kernel<4><<<dim3(DMODEL / 64, MGRID), blk, 0, stream>>>(
        ygated, DINNER, W_out, DMODEL, preln, DMODEL, DINNER, 0);

    // 7) LayerNorm
    layernorm_kernel<<<BB * LL, blk, 0, stream>>>(preln, ln_g, ln_b, out);
}